// EmbeddingLayer_35948876268305
// MI455X (gfx1250) — compile-verified
//
#include <hip/hip_runtime.h>
#include <math.h>

#define B_    2
#define L_    256
#define S_    384
#define H_    768
#define Tc_   16
#define Dc_   50
#define WD_   100
#define OUTW  (H_ + WD_ + 2*Dc_)   // 968
#define G4    (4*Dc_)              // 200 gates
#define KPAD  52                   // Dc padded to 13 chunks of K=4
#define NPAD  208                  // gates padded to 13 tiles of 16
#define NTILES 13
#define LSTM_THREADS (NTILES*32)   // 13 waves (wave32)
#define NEGINF (-3.4028235e38f)

typedef __attribute__((ext_vector_type(2))) float v2f;
typedef __attribute__((ext_vector_type(8))) float v8f;

__device__ __forceinline__ float sigmoidf_(float x) {
    return 1.0f / (1.0f + __expf(-x));
}

// ---------- global min of bert_emb (2-stage) ----------
__global__ void kmin_partial(const float* __restrict__ x, int n, float* __restrict__ part) {
    __shared__ float red[256];
    float m = 3.4028235e38f;
    for (int i = blockIdx.x * blockDim.x + threadIdx.x; i < n; i += gridDim.x * blockDim.x)
        m = fminf(m, x[i]);
    red[threadIdx.x] = m;
    __syncthreads();
    for (int off = 128; off > 0; off >>= 1) {
        if (threadIdx.x < off) red[threadIdx.x] = fminf(red[threadIdx.x], red[threadIdx.x + off]);
        __syncthreads();
    }
    if (threadIdx.x == 0) part[blockIdx.x] = red[0];
}

__global__ void kmin_final(const float* __restrict__ part, float* __restrict__ out) {
    __shared__ float red[256];
    red[threadIdx.x] = part[threadIdx.x];
    __syncthreads();
    for (int off = 128; off > 0; off >>= 1) {
        if (threadIdx.x < off) red[threadIdx.x] = fminf(red[threadIdx.x], red[threadIdx.x + off]);
        __syncthreads();
    }
    if (threadIdx.x == 0) out[0] = red[0];
}

// ---------- word_reps: masked max over pieces (L2-resident bert_emb) ----------
__global__ void kwordreps(const float* __restrict__ bert, const int* __restrict__ p2w,
                          const float* __restrict__ minv_p, float* __restrict__ out) {
    __shared__ int mask[S_];
    const int w = blockIdx.x;          // b*L + l
    const int b = w / L_;
    const int* mrow = p2w + w * S_;
    for (int s = threadIdx.x; s < S_; s += blockDim.x) mask[s] = mrow[s];
    __syncthreads();

    const int t = threadIdx.x;         // 256 threads, 3 H-columns each
    const float* base = bert + (size_t)b * S_ * H_;
    float m0 = NEGINF, m1 = NEGINF, m2 = NEGINF;
    bool any = false;
    for (int s = 0; s < S_; ++s) {
        if (mask[s]) {                 // uniform across the block -> scalar branch
            any = true;
            const float* p = base + s * H_;
            m0 = fmaxf(m0, p[t]);
            m1 = fmaxf(m1, p[t + 256]);
            m2 = fmaxf(m2, p[t + 512]);
        }
    }
    if (!any) { float mv = minv_p[0]; m0 = mv; m1 = mv; m2 = mv; }
    float* orow = out + (size_t)w * OUTW;
    orow[t] = m0; orow[t + 256] = m1; orow[t + 512] = m2;
}

// ---------- word embedding gather ----------
__global__ void kembed(const int* __restrict__ wid, const float* __restrict__ table,
                       float* __restrict__ out) {
    int i = blockIdx.x * blockDim.x + threadIdx.x;
    if (i >= B_ * L_ * WD_) return;
    int w = i / WD_, d = i - w * WD_;
    out[(size_t)w * OUTW + H_ + d] = table[(size_t)wid[w] * WD_ + d];
}

// ---------- char bi-LSTM via V_WMMA_F32_16X16X4_F32, running max over valid t ----------
__global__ void __launch_bounds__(LSTM_THREADS) klstm(
    const int* __restrict__ char_count, const int* __restrict__ char_ids,
    const float* __restrict__ char_table,
    const float* __restrict__ Wih_f, const float* __restrict__ Whh_f,
    const float* __restrict__ bih_f, const float* __restrict__ bhh_f,
    const float* __restrict__ Wih_b, const float* __restrict__ Whh_b,
    const float* __restrict__ bih_b, const float* __restrict__ bhh_b,
    float* __restrict__ out)
{
    const int dir = blockIdx.y;
    const float* Wih = dir ? Wih_b : Wih_f;
    const float* Whh = dir ? Whh_b : Whh_f;
    const float* bih = dir ? bih_b : bih_f;
    const float* bhh = dir ? bhh_b : bhh_f;

    __shared__ float x_lds[16][KPAD];   // x_t, K-padded (cols 50..51 stay 0)
    __shared__ float h_lds[16][KPAD];   // h_{t-1}
    __shared__ float g_buf[16][NPAD];   // gate pre-activations
    __shared__ float c_st[16][Dc_];
    __shared__ float hmax[16][Dc_];
    __shared__ float bias_lds[NPAD];
    __shared__ int   len_lds[16];

    const int tid   = threadIdx.x;
    const int wave  = tid >> 5;
    const int lane  = tid & 31;
    const int mrow  = lane & 15;        // A-matrix M index / C-N index
    const int khalf = lane >> 4;        // selects K pair (A/B) and M half (C/D)
    const int ncol  = wave * 16 + mrow; // this wave's gate column for B/C

    // ---- init state ----
    for (int i = tid; i < 16 * KPAD; i += LSTM_THREADS) {
        ((float*)x_lds)[i] = 0.0f;
        ((float*)h_lds)[i] = 0.0f;
    }
    for (int i = tid; i < 16 * Dc_; i += LSTM_THREADS) {
        ((float*)c_st)[i] = 0.0f;
        ((float*)hmax)[i] = NEGINF;
    }
    for (int n = tid; n < NPAD; n += LSTM_THREADS)
        bias_lds[n] = (n < G4) ? (bih[n] + bhh[n]) : 0.0f;
    if (tid < 16) {
        int cc = char_count[blockIdx.x * 16 + tid];
        len_lds[tid] = cc > 1 ? cc : 1;
    }

    // ---- time-invariant weight B-fragments, resident in VGPRs for all 16 steps ----
    v2f bIh[NTILES], bHh[NTILES];
    #pragma unroll
    for (int kc = 0; kc < NTILES; ++kc) {
        int k0 = kc * 4 + 2 * khalf;
        bool nok = (ncol < G4);
        bIh[kc].x = (nok && k0     < Dc_) ? Wih[ncol * Dc_ + k0]     : 0.0f;
        bIh[kc].y = (nok && k0 + 1 < Dc_) ? Wih[ncol * Dc_ + k0 + 1] : 0.0f;
        bHh[kc].x = (nok && k0     < Dc_) ? Whh[ncol * Dc_ + k0]     : 0.0f;
        bHh[kc].y = (nok && k0 + 1 < Dc_) ? Whh[ncol * Dc_ + k0 + 1] : 0.0f;
    }
    __syncthreads();

    for (int t = 0; t < Tc_; ++t) {
        // stage x_t (backward dir gathers reversed-within-length position)
        for (int e = tid; e < 16 * Dc_; e += LSTM_THREADS) {
            int s = e / Dc_, k = e - s * Dc_;
            int len = len_lds[s];
            int pos = (dir == 0) ? t : ((t < len) ? (len - 1 - t) : t);
            int gseq = blockIdx.x * 16 + s;
            int cid = char_ids[gseq * Tc_ + pos];
            x_lds[s][k] = char_table[cid * Dc_ + k];
        }
        __syncthreads();

        // g = x_t @ Wih^T + h @ Whh^T : one 16(seq) x 16(gate) tile per wave
        v8f acc = {};
        #pragma unroll
        for (int kc = 0; kc < NTILES; ++kc) {
            int kb = kc * 4 + 2 * khalf;
            v2f aX; aX.x = x_lds[mrow][kb]; aX.y = x_lds[mrow][kb + 1];
            acc = __builtin_amdgcn_wmma_f32_16x16x4_f32(
                false, aX, false, bIh[kc], (short)0, acc, false, false);
            v2f aH; aH.x = h_lds[mrow][kb]; aH.y = h_lds[mrow][kb + 1];
            acc = __builtin_amdgcn_wmma_f32_16x16x4_f32(
                false, aH, false, bHh[kc], (short)0, acc, false, false);
        }
        #pragma unroll
        for (int r = 0; r < 8; ++r) {
            int srow = r + 8 * khalf;       // C/D layout: M = r + 8*(lane>=16)
            g_buf[srow][ncol] = acc[r];
        }
        __syncthreads();

        // LSTM cell elementwise + running max over valid steps
        for (int e = tid; e < 16 * Dc_; e += LSTM_THREADS) {
            int s = e / Dc_, j = e - s * Dc_;
            float gi = g_buf[s][j]           + bias_lds[j];
            float gf = g_buf[s][Dc_ + j]     + bias_lds[Dc_ + j];
            float gg = g_buf[s][2 * Dc_ + j] + bias_lds[2 * Dc_ + j];
            float go = g_buf[s][3 * Dc_ + j] + bias_lds[3 * Dc_ + j];
            float iv = sigmoidf_(gi);
            float fv = sigmoidf_(gf);
            float gv = tanhf(gg);
            float ov = sigmoidf_(go);
            float c  = fv * c_st[s][j] + iv * gv;
            c_st[s][j] = c;
            float h = ov * tanhf(c);
            h_lds[s][j] = h;
            if (t < len_lds[s]) hmax[s][j] = fmaxf(hmax[s][j], h);
        }
        __syncthreads();
    }

    // sup output: fwd -> cols [868..917], bwd -> cols [918..967]
    for (int e = tid; e < 16 * Dc_; e += LSTM_THREADS) {
        int s = e / Dc_, j = e - s * Dc_;
        int gseq = blockIdx.x * 16 + s;
        out[(size_t)gseq * OUTW + H_ + WD_ + dir * Dc_ + j] = hmax[s][j];
    }
}

extern "C" void kernel_launch(void* const* d_in, const int* in_sizes, int n_in,
                              void* d_out, int out_size, void* d_ws, size_t ws_size,
                              hipStream_t stream) {
    (void)in_sizes; (void)n_in; (void)out_size; (void)ws_size;
    const float* bert   = (const float*)d_in[0];
    const int*   p2w    = (const int*)d_in[1];
    const int*   wid    = (const int*)d_in[2];
    const int*   ccount = (const int*)d_in[3];
    const int*   cids   = (const int*)d_in[4];
    /* d_in[5] token_masks_char: implied by char_count (>=1), unused */
    const float* wtab   = (const float*)d_in[6];
    const float* ctab   = (const float*)d_in[7];
    const float* Wih_f  = (const float*)d_in[8];
    const float* Whh_f  = (const float*)d_in[9];
    const float* bih_f  = (const float*)d_in[10];
    const float* bhh_f  = (const float*)d_in[11];
    const float* Wih_b  = (const float*)d_in[12];
    const float* Whh_b  = (const float*)d_in[13];
    const float* bih_b  = (const float*)d_in[14];
    const float* bhh_b  = (const float*)d_in[15];
    float* out = (float*)d_out;
    float* wsf = (float*)d_ws;   // [0..255] partial mins, [256] global min

    kmin_partial<<<256, 256, 0, stream>>>(bert, B_ * S_ * H_, wsf);
    kmin_final<<<1, 256, 0, stream>>>(wsf, wsf + 256);
    kwordreps<<<B_ * L_, 256, 0, stream>>>(bert, p2w, wsf + 256, out);
    kembed<<<(B_ * L_ * WD_ + 255) / 256, 256, 0, stream>>>(wid, wtab, out);
    klstm<<<dim3(B_ * L_ / 16, 2), LSTM_THREADS, 0, stream>>>(
        ccount, cids, ctab,
        Wih_f, Whh_f, bih_f, bhh_f,
        Wih_b, Whh_b, bih_b, bhh_b, out);
}